// EEACSF_22471268893093
// MI455X (gfx1250) — compile-verified
//
#include <hip/hip_runtime.h>
#include <hip/hip_bf16.h>

// ---------------- problem constants (from reference) ----------------
#define OUT_COLS   407
#define ONEHOT_N   87
#define RAD_OFF    87      // radial cols start (64 cols)
#define ANG_OFF    151     // angular cols start (256 cols)
#define RDIV       16
// radial : shiftR[i] = 0.8 + 0.275*i, eta=16, coeff 0.25
// angular: shiftA[a] = 0.8 + 0.675*a (eta=8), shiftZ[s] = pi/8 + s*pi/4, zeta=32

typedef float v2f __attribute__((ext_vector_type(2)));
typedef float v8f __attribute__((ext_vector_type(8)));

// ---------------- init: zero first ncols, set onehot -----------------------
__global__ void k_init_out(const int* __restrict__ species,
                           float* __restrict__ out, int n_atoms, int ncols) {
    int idx = blockIdx.x * blockDim.x + threadIdx.x;
    int total = n_atoms * ncols;
    if (idx >= total) return;
    int n = idx / ncols;
    int c = idx - n * ncols;
    float v = 0.0f;
    if (c < ONEHOT_N && c == species[n]) v = 1.0f;
    out[(size_t)n * OUT_COLS + c] = v;
}

// ---------------- histogram (generic over key array) -----------------------
__global__ void k_hist(const int* __restrict__ key, int* __restrict__ counts,
                       int n) {
    int p = blockIdx.x * blockDim.x + threadIdx.x;
    if (p >= n) return;
    atomicAdd(&counts[key[p]], 1);
}

// ---------------- single-block exclusive scan ------------------------------
__global__ void __launch_bounds__(1024)
k_scan(const int* __restrict__ counts, int* __restrict__ offsets,
       int* __restrict__ cursor, int n) {
    __shared__ int sm[1024];
    __shared__ int carry;
    const int tid = threadIdx.x;
    if (tid == 0) { carry = 0; offsets[0] = 0; }
    __syncthreads();
    for (int base = 0; base < n; base += 1024) {
        int i = base + tid;
        int v = (i < n) ? counts[i] : 0;
        sm[tid] = v;
        __syncthreads();
        for (int off = 1; off < 1024; off <<= 1) {     // Hillis-Steele inclusive
            int t = (tid >= off) ? sm[tid - off] : 0;
            __syncthreads();
            sm[tid] += t;
            __syncthreads();
        }
        int inc = sm[tid];
        int c = carry;
        int tot = sm[1023];
        __syncthreads();
        if (i < n) {
            offsets[i + 1] = c + inc;
            cursor[i]      = c + inc - v;
        }
        if (tid == 1023) carry = c + tot;
        __syncthreads();
    }
}

// ---------------- radial: scatter packed edge records ----------------------
__global__ void k_scatter_r(const float* __restrict__ dist,
                            const float* __restrict__ sw,
                            const int* __restrict__ esrc,
                            const int* __restrict__ edst,
                            const int* __restrict__ species,
                            int* __restrict__ cursor,
                            float4* __restrict__ buf, int n_edges) {
    int e = blockIdx.x * blockDim.x + threadIdx.x;
    if (e >= n_edges) return;
    int pos = atomicAdd(&cursor[esrc[e]], 1);
    float4 rec;
    rec.x = dist[e];
    rec.y = sw[e];
    rec.z = __uint_as_float((unsigned)species[edst[e]]);
    rec.w = 0.0f;
    buf[pos] = rec;
}

__device__ __forceinline__ int iclamp(int p, int emax) {
    return (p < emax) ? p : emax;
}

// ---------------- radial: per-atom 16x4 via WMMA 16x16x4 f32 ---------------
// A[m,k] = 0.25*exp(-16*(d_k - shiftR_m)^2)*sw_k   (16 x 4)
// B[k,n] = valence_table[sp_k][n] for n<4 else 0   (4 x 16, 12 cols wasted)
// Fully branchless: record index clamped to e-1 (valid); A masked to 0 for
// k>=e; B column clamped to m&3 (always a valid table slot) and masked for m>=4.
__global__ void __launch_bounds__(256)
k_radial_wmma(const int* __restrict__ offsets,
              const float4* __restrict__ buf,
              const float* __restrict__ vt,
              float* __restrict__ out, int n_atoms) {
    const int lane = threadIdx.x & 31;
    const int wv   = threadIdx.x >> 5;
    const int atom = blockIdx.x * (blockDim.x >> 5) + wv;
    if (atom >= n_atoms) return;
    const int s = offsets[atom];
    const int e = offsets[atom + 1];
    const int half = lane >> 4;
    const int m    = lane & 15;
    const int mc   = m & 3;          // clamped B column (valid for all lanes)
    const float shiftR = 0.8f + 0.275f * (float)m;

    v8f c0 = {0.f, 0.f, 0.f, 0.f, 0.f, 0.f, 0.f, 0.f};
    v8f c1 = c0;
    for (int base = s; base < e; base += 8) {
        v2f a0, b0, a1, b1;
#pragma unroll
        for (int k = 0; k < 2; ++k) {
            int p0 = base + 2 * half + k;
            int p1 = p0 + 4;
            float4 r0 = buf[iclamp(p0, e - 1)];
            float4 r1 = buf[iclamp(p1, e - 1)];
            float x0 = r0.x - shiftR;
            float x1 = r1.x - shiftR;
            float av0 = 0.25f * __expf(-16.0f * x0 * x0) * r0.y;
            float av1 = 0.25f * __expf(-16.0f * x1 * x1) * r1.y;
            av0 = (p0 < e) ? av0 : 0.0f;
            av1 = (p1 < e) ? av1 : 0.0f;
            float t0 = vt[__float_as_uint(r0.z) * 4 + mc];   // unconditional load
            float t1 = vt[__float_as_uint(r1.z) * 4 + mc];
            float bv0 = (m < 4) ? t0 : 0.0f;                 // value mask only
            float bv1 = (m < 4) ? t1 : 0.0f;
            if (k == 0) { a0.x = av0; b0.x = bv0; a1.x = av1; b1.x = bv1; }
            else        { a0.y = av0; b0.y = bv0; a1.y = av1; b1.y = bv1; }
        }
        c0 = __builtin_amdgcn_wmma_f32_16x16x4_f32(false, a0, false, b0,
                                                   (short)0, c0, false, false);
        c1 = __builtin_amdgcn_wmma_f32_16x16x4_f32(false, a1, false, b1,
                                                   (short)0, c1, false, false);
    }
    if (m < 4) {
        float* o = out + (size_t)atom * OUT_COLS + RAD_OFF + (half * 8) * 4 + m;
#pragma unroll
        for (int r = 0; r < 8; ++r) o[r * 4] = c0[r] + c1[r];
    }
}

// ---------------- radial fallback: direct float atomics --------------------
__global__ void k_radial_atomic(const float* __restrict__ dist,
                                const float* __restrict__ sw,
                                const int* __restrict__ esrc,
                                const int* __restrict__ edst,
                                const int* __restrict__ species,
                                const float* __restrict__ vt,
                                float* __restrict__ out, int n_edges) {
    int idx = blockIdx.x * blockDim.x + threadIdx.x;
    if (idx >= n_edges * RDIV) return;
    int e = idx >> 4;
    int i = idx & 15;
    float x = dist[e] - (0.8f + 0.275f * (float)i);
    float t = 0.25f * __expf(-16.0f * x * x) * sw[e];
    const float* v = vt + 4 * species[edst[e]];
    float* o = out + (size_t)esrc[e] * OUT_COLS + RAD_OFF + i * 4;
    unsafeAtomicAdd(o + 0, t * v[0]);
    unsafeAtomicAdd(o + 1, t * v[1]);
    unsafeAtomicAdd(o + 2, t * v[2]);
    unsafeAtomicAdd(o + 3, t * v[3]);
}

// ---------------- angular: scatter packed pair records ---------------------
__global__ void k_scatter_a(const float* __restrict__ ang_angles,
                            const float* __restrict__ ang_dist,
                            const float* __restrict__ ang_sw,
                            const int* __restrict__ ang_edge_dst,
                            const int* __restrict__ angle_src,
                            const int* __restrict__ angle_dst,
                            const int* __restrict__ central,
                            const int* __restrict__ species,
                            int* __restrict__ cursor,
                            float4* __restrict__ buf, int n_pairs) {
    int p = blockIdx.x * blockDim.x + threadIdx.x;
    if (p >= n_pairs) return;
    int e1 = angle_src[p];
    int e2 = angle_dst[p];
    unsigned sp1 = (unsigned)species[ang_edge_dst[e1]];
    unsigned sp2 = (unsigned)species[ang_edge_dst[e2]];
    int pos = atomicAdd(&cursor[central[p]], 1);
    float4 rec;
    rec.x = ang_angles[p];
    rec.y = 0.5f * (ang_dist[e1] + ang_dist[e2]);
    rec.z = 2.0f * ang_sw[e1] * ang_sw[e2];
    rec.w = __uint_as_float((sp1 << 16) | sp2);
    buf[pos] = rec;
}

__device__ __forceinline__ float ang_term(float4 pb, float shiftA, float shiftZ) {
    float xd = pb.y - shiftA;
    float f2 = __expf(-8.0f * xd * xd);
    float cb = 0.5f + 0.5f * __cosf(pb.x - shiftZ);   // |arg| <= pi, fast path ok
    cb = cb * cb; cb = cb * cb; cb = cb * cb;
    cb = cb * cb; cb = cb * cb;                       // ^32
    return f2 * cb * pb.z;
}

// ---------------- angular: per-atom 16x16 via WMMA 16x16x4 f32 -------------
__global__ void __launch_bounds__(256)
k_angular_wmma(const int* __restrict__ offsets,
               const float4* __restrict__ buf,
               const float* __restrict__ vt,
               float* __restrict__ out, int n_atoms) {
    const int lane = threadIdx.x & 31;
    const int wv   = threadIdx.x >> 5;
    const int atom = blockIdx.x * (blockDim.x >> 5) + wv;
    if (atom >= n_atoms) return;
    const int s = offsets[atom];
    const int e = offsets[atom + 1];
    const int half = lane >> 4;
    const int m    = lane & 15;
    const int hi   = m >> 2;   // A: dist idx      B: vp idx
    const int lo   = m & 3;    // A: section idx   B: vm idx
    const float shiftA = 0.8f + 0.675f * (float)hi;
    const float shiftZ = 0.39269908169872414f + 0.7853981633974483f * (float)lo;

    v8f c0 = {0.f, 0.f, 0.f, 0.f, 0.f, 0.f, 0.f, 0.f};
    v8f c1 = c0;
    for (int base = s; base < e; base += 8) {
        v2f a0, b0, a1, b1;
#pragma unroll
        for (int k = 0; k < 2; ++k) {
            int p0 = base + 2 * half + k;
            int p1 = p0 + 4;
            float4 r0 = buf[iclamp(p0, e - 1)];
            float4 r1 = buf[iclamp(p1, e - 1)];
            float av0 = ang_term(r0, shiftA, shiftZ);
            float av1 = ang_term(r1, shiftA, shiftZ);
            av0 = (p0 < e) ? av0 : 0.0f;
            av1 = (p1 < e) ? av1 : 0.0f;
            unsigned s0 = __float_as_uint(r0.w);
            unsigned s1 = __float_as_uint(r1.w);
            const float* q0a = vt + 4 * (s0 >> 16);
            const float* q0b = vt + 4 * (s0 & 0xffffu);
            const float* q1a = vt + 4 * (s1 >> 16);
            const float* q1b = vt + 4 * (s1 & 0xffffu);
            float bv0 = (q0a[hi] + q0b[hi]) * (q0a[lo] * q0b[lo]);
            float bv1 = (q1a[hi] + q1b[hi]) * (q1a[lo] * q1b[lo]);
            if (k == 0) { a0.x = av0; b0.x = bv0; a1.x = av1; b1.x = bv1; }
            else        { a0.y = av0; b0.y = bv0; a1.y = av1; b1.y = bv1; }
        }
        c0 = __builtin_amdgcn_wmma_f32_16x16x4_f32(false, a0, false, b0,
                                                   (short)0, c0, false, false);
        c1 = __builtin_amdgcn_wmma_f32_16x16x4_f32(false, a1, false, b1,
                                                   (short)0, c1, false, false);
    }
    float* o = out + (size_t)atom * OUT_COLS + ANG_OFF + (half * 8) * 16 + m;
#pragma unroll
    for (int r = 0; r < 8; ++r) o[r * 16] = c0[r] + c1[r];
}

// ---------------- angular fallback: direct float atomics -------------------
__global__ void k_angular_atomic(const float* __restrict__ ang_angles,
                                 const float* __restrict__ ang_dist,
                                 const float* __restrict__ ang_sw,
                                 const int* __restrict__ ang_edge_dst,
                                 const int* __restrict__ angle_src,
                                 const int* __restrict__ angle_dst,
                                 const int* __restrict__ central,
                                 const int* __restrict__ species,
                                 const float* __restrict__ vt,
                                 float* __restrict__ out, int n_pairs) {
    int idx = blockIdx.x * blockDim.x + threadIdx.x;
    if (idx >= n_pairs * 16) return;
    int p = idx >> 4;
    int m = idx & 15;
    int hi = m >> 2, lo = m & 3;
    int e1 = angle_src[p], e2 = angle_dst[p];
    float d12 = 0.5f * (ang_dist[e1] + ang_dist[e2]);
    float xd = d12 - (0.8f + 0.675f * (float)hi);
    float f2 = __expf(-8.0f * xd * xd);
    float cb = 0.5f + 0.5f * __cosf(ang_angles[p] -
                 (0.39269908169872414f + 0.7853981633974483f * (float)lo));
    cb = cb * cb; cb = cb * cb; cb = cb * cb; cb = cb * cb; cb = cb * cb;
    float term = f2 * cb * 2.0f * ang_sw[e1] * ang_sw[e2];
    const float* r1 = vt + 4 * species[ang_edge_dst[e1]];
    const float* r2 = vt + 4 * species[ang_edge_dst[e2]];
    float* o = out + (size_t)central[p] * OUT_COLS + ANG_OFF + m * 16;
#pragma unroll
    for (int n = 0; n < 16; ++n)
        unsafeAtomicAdd(o + n, term * (r1[n >> 2] + r2[n >> 2]) *
                                  (r1[n & 3] * r2[n & 3]));
}

// ---------------- launcher -------------------------------------------------
extern "C" void kernel_launch(void* const* d_in, const int* in_sizes, int n_in,
                              void* d_out, int out_size, void* d_ws, size_t ws_size,
                              hipStream_t stream) {
    const int*   species      = (const int*)  d_in[0];
    const float* distances    = (const float*)d_in[1];
    const float* eswitch      = (const float*)d_in[2];
    const int*   edge_src     = (const int*)  d_in[3];
    const int*   edge_dst     = (const int*)  d_in[4];
    const float* ang_angles   = (const float*)d_in[5];
    const float* ang_dist     = (const float*)d_in[6];
    const float* ang_sw       = (const float*)d_in[7];
    const int*   ang_edge_dst = (const int*)  d_in[8];
    const int*   angle_src    = (const int*)  d_in[9];
    const int*   angle_dst    = (const int*)  d_in[10];
    const int*   central      = (const int*)  d_in[11];
    const float* vt           = (const float*)d_in[12];

    const int n_atoms = in_sizes[0];
    const int n_edges = in_sizes[1];
    const int n_pairs = in_sizes[5];
    float* out = (float*)d_out;

    // workspace: int counters + one shared record buffer (reused radial/angular)
    char* ws = (char*)d_ws;
    int* counts  = (int*)ws;
    int* offsets = counts + n_atoms;
    int* cursor  = offsets + n_atoms + 1;
    size_t hdr = ((size_t)(3 * n_atoms + 1) * sizeof(int) + 255) & ~(size_t)255;
    int maxrec = (n_edges > n_pairs) ? n_edges : n_pairs;
    size_t needed = hdr + (size_t)maxrec * sizeof(float4);
    float4* buf = (float4*)(ws + hdr);
    const bool sorted_path = (ws_size >= needed);

    const int waves_per_block = 256 / 32;
    const int atom_blocks = (n_atoms + waves_per_block - 1) / waves_per_block;

    if (sorted_path) {
        // init onehot only; radial+angular regions fully overwritten densely
        int total = n_atoms * ONEHOT_N;
        k_init_out<<<(total + 255) / 256, 256, 0, stream>>>(species, out,
                                                            n_atoms, ONEHOT_N);
        // ----- radial phase -----
        hipMemsetAsync(counts, 0, (size_t)n_atoms * sizeof(int), stream);
        k_hist<<<(n_edges + 255) / 256, 256, 0, stream>>>(edge_src, counts, n_edges);
        k_scan<<<1, 1024, 0, stream>>>(counts, offsets, cursor, n_atoms);
        k_scatter_r<<<(n_edges + 255) / 256, 256, 0, stream>>>(
            distances, eswitch, edge_src, edge_dst, species, cursor, buf, n_edges);
        k_radial_wmma<<<atom_blocks, 256, 0, stream>>>(offsets, buf, vt, out, n_atoms);
        // ----- angular phase (reuses counters + buf) -----
        hipMemsetAsync(counts, 0, (size_t)n_atoms * sizeof(int), stream);
        k_hist<<<(n_pairs + 255) / 256, 256, 0, stream>>>(central, counts, n_pairs);
        k_scan<<<1, 1024, 0, stream>>>(counts, offsets, cursor, n_atoms);
        k_scatter_a<<<(n_pairs + 255) / 256, 256, 0, stream>>>(
            ang_angles, ang_dist, ang_sw, ang_edge_dst, angle_src, angle_dst,
            central, species, cursor, buf, n_pairs);
        k_angular_wmma<<<atom_blocks, 256, 0, stream>>>(offsets, buf, vt, out, n_atoms);
    } else {
        // atomic fallback: zero entire output rows, then scatter-accumulate
        int total = n_atoms * OUT_COLS;
        k_init_out<<<(total + 255) / 256, 256, 0, stream>>>(species, out,
                                                            n_atoms, OUT_COLS);
        k_radial_atomic<<<(n_edges * RDIV + 255) / 256, 256, 0, stream>>>(
            distances, eswitch, edge_src, edge_dst, species, vt, out, n_edges);
        k_angular_atomic<<<(n_pairs * 16 + 255) / 256, 256, 0, stream>>>(
            ang_angles, ang_dist, ang_sw, ang_edge_dst, angle_src, angle_dst,
            central, species, vt, out, n_pairs);
    }
}